// GCNConv_87514253623560
// MI455X (gfx1250) — compile-verified
//
#include <hip/hip_runtime.h>
#include <hip/hip_bf16.h>
#include <stdint.h>

// ---------------------------------------------------------------------------
// GCN scatter-sum: out[src] += feat[dst] * w    (N=100000 nodes, E=1M, D=64)
// Primary: CSR build over `src` (histogram -> scan -> payload permute), then
// one wave32 per node accumulates in registers (2 floats/lane, no fp atomics).
// Per-node (dst,w) chunks are staged to LDS with gfx1250 async global->LDS
// loads (ASYNCcnt). Fallback (ws too small): tiled scatter with inline-asm
// global_atomic_add_f32.
// ---------------------------------------------------------------------------

#define D_FEAT 64
#define TILE   256   // edges per block-tile in atomic fallback path

// gfx1250 async global->LDS load, GVS mode: mem = SGPR64 base + VGPR32 offset
__device__ __forceinline__ void async_g2l_b32(uint32_t lds_off, uint32_t voff,
                                              uint64_t sbase) {
    asm volatile("global_load_async_to_lds_b32 %0, %1, %2"
                 :: "v"(lds_off), "v"(voff), "s"(sbase)
                 : "memory");
}
__device__ __forceinline__ void wait_asynccnt0() {
    asm volatile("s_wait_asynccnt 0x0" ::: "memory");
}
// non-returning hardware fp32 atomic add (STOREcnt-tracked)
__device__ __forceinline__ void atomic_add_f32_hw(uint64_t sbase, uint32_t voff,
                                                  float v) {
    asm volatile("global_atomic_add_f32 %0, %1, %2"
                 :: "v"(voff), "v"(v), "s"(sbase)
                 : "memory");
}
// low 32 bits of a flat pointer into LDS == LDS byte address
__device__ __forceinline__ uint32_t lds_addr(const void* p) {
    return (uint32_t)(uintptr_t)p;
}

// ---------------------------------------------------------------------------
__global__ __launch_bounds__(256) void k_zero_u32(unsigned* p, size_t n) {
    for (size_t i = (size_t)blockIdx.x * blockDim.x + threadIdx.x; i < n;
         i += (size_t)gridDim.x * blockDim.x)
        p[i] = 0u;
}

__global__ __launch_bounds__(256) void k_hist(const int* __restrict__ src,
                                              unsigned* __restrict__ cnt, int E) {
    for (int e = blockIdx.x * blockDim.x + threadIdx.x; e < E;
         e += gridDim.x * blockDim.x)
        atomicAdd(&cnt[src[e]], 1u);
}

// single-workgroup exclusive scan over cnt[0..N-1] -> off[], cur[]; off[N]=E
__global__ __launch_bounds__(1024) void k_scan(const unsigned* __restrict__ cnt,
                                               unsigned* __restrict__ off,
                                               unsigned* __restrict__ cur, int N) {
    __shared__ unsigned s[1024];
    __shared__ unsigned s_run;
    const int tid = threadIdx.x;
    if (tid == 0) s_run = 0u;
    __syncthreads();
    for (int base = 0; base < N; base += 1024) {
        int i = base + tid;
        unsigned v = (i < N) ? cnt[i] : 0u;
        s[tid] = v;
        __syncthreads();
        for (int o = 1; o < 1024; o <<= 1) {       // Hillis-Steele inclusive
            unsigned t = (tid >= o) ? s[tid - o] : 0u;
            __syncthreads();
            s[tid] += t;
            __syncthreads();
        }
        unsigned incl = s[tid];
        unsigned excl = s_run + incl - v;
        if (i < N) { off[i] = excl; cur[i] = excl; }
        __syncthreads();                 // everyone done reading s_run
        if (tid == 1023) s_run += incl;  // incl of last lane == chunk total
        __syncthreads();
    }
    if (tid == 0) off[N] = s_run;
}

// permute the edge PAYLOAD (dst, w) into CSR order -> gather kernel streams
// contiguous ranges with no second indirection
__global__ __launch_bounds__(256) void k_perm(const int* __restrict__ src,
                                              const int* __restrict__ dst,
                                              const float* __restrict__ ew,
                                              unsigned* __restrict__ cur,
                                              int* __restrict__ dstS,
                                              float* __restrict__ wS, int E) {
    for (int e = blockIdx.x * blockDim.x + threadIdx.x; e < E;
         e += gridDim.x * blockDim.x) {
        unsigned p = atomicAdd(&cur[src[e]], 1u);
        dstS[p] = dst[e];
        wS[p]   = ew[e];
    }
}

// one wave32 per node; lane holds features {2*lane, 2*lane+1}; no fp atomics.
// (dst,w) chunks staged to this wave's LDS slice via async global->LDS loads.
__global__ __launch_bounds__(256) void k_gather(const float* __restrict__ feat,
                                                const float* __restrict__ wS,
                                                const int* __restrict__ dstS,
                                                const unsigned* __restrict__ off,
                                                float* __restrict__ out, int N) {
    __shared__ int   s_d[8 * 32];
    __shared__ float s_w[8 * 32];
    const int wv   = threadIdx.x >> 5;
    const int lane = threadIdx.x & 31;
    const int wpb  = blockDim.x >> 5;
    const uint64_t bdst = (uint64_t)(uintptr_t)dstS;
    const uint64_t bw   = (uint64_t)(uintptr_t)wS;
    const float2* __restrict__ feat2 = (const float2*)feat;

    for (int node = blockIdx.x * wpb + wv; node < N; node += gridDim.x * wpb) {
        unsigned b = off[node], e = off[node + 1];
        float a0 = 0.f, a1 = 0.f;
        for (unsigned c = b; c < e; c += 32u) {
            int m = (int)((e - c) < 32u ? (e - c) : 32u);
            if (lane < m) {
                uint32_t vo = (c + (unsigned)lane) * 4u;   // byte offset
                async_g2l_b32(lds_addr(&s_d[wv * 32 + lane]), vo, bdst);
                async_g2l_b32(lds_addr(&s_w[wv * 32 + lane]), vo, bw);
            }
            wait_asynccnt0();   // this wave's LDS slice is ready
            for (int j = 0; j < m; ++j) {
                int   dj = s_d[wv * 32 + j];   // LDS broadcast read
                float wj = s_w[wv * 32 + j];
                float2 f = feat2[(size_t)dj * 32 + lane];
                a0 += wj * f.x;
                a1 += wj * f.y;
            }
        }
        ((float2*)out)[(size_t)node * 32 + lane] = make_float2(a0, a1);
    }
}

// ---------------------------------------------------------------------------
// Fallback: tiled atomic scatter. Edge tiles staged to LDS with async loads;
// each wave cooperates on one 256B feature row per edge; hardware fp32
// atomics via inline asm (never a CAS loop).
// ---------------------------------------------------------------------------
__global__ __launch_bounds__(256) void k_scatter_atomic(
        const float* __restrict__ feat, const float* __restrict__ ew,
        const int* __restrict__ src, const int* __restrict__ dst,
        float* __restrict__ out, int E) {
    __shared__ int   s_src[TILE];
    __shared__ int   s_dst[TILE];
    __shared__ float s_w[TILE];
    const int tid  = threadIdx.x;
    const int wv   = tid >> 5;
    const int lane = tid & 31;
    const uint64_t bout = (uint64_t)(uintptr_t)out;
    const float2* __restrict__ feat2 = (const float2*)feat;

    for (long long base = (long long)blockIdx.x * TILE; base < E;
         base += (long long)gridDim.x * TILE) {
        int n = (int)(((long long)E - base) < TILE ? ((long long)E - base) : TILE);
        if (tid < n) {
            uint32_t vo = (uint32_t)((base + tid) * 4ll);
            async_g2l_b32(lds_addr(&s_src[tid]), vo, (uint64_t)(uintptr_t)src);
            async_g2l_b32(lds_addr(&s_dst[tid]), vo, (uint64_t)(uintptr_t)dst);
            async_g2l_b32(lds_addr(&s_w[tid]),   vo, (uint64_t)(uintptr_t)ew);
        }
        wait_asynccnt0();     // my wave's async LDS writes are done
        __syncthreads();      // everyone's are visible

        int cnt = n - wv * 32;
        cnt = cnt < 0 ? 0 : (cnt > 32 ? 32 : cnt);
        for (int j = 0; j < cnt; ++j) {
            int   i = wv * 32 + j;
            int   s = s_src[i];
            int   d = s_dst[i];
            float w = s_w[i];
            float2 f = feat2[(size_t)d * 32 + lane];
            uint32_t vo = (uint32_t)(((unsigned)s * D_FEAT + lane * 2) * 4u);
            atomic_add_f32_hw(bout, vo,     w * f.x);
            atomic_add_f32_hw(bout, vo + 4, w * f.y);
        }
        __syncthreads();      // before next tile overwrites LDS
    }
}

// ---------------------------------------------------------------------------
extern "C" void kernel_launch(void* const* d_in, const int* in_sizes, int n_in,
                              void* d_out, int out_size, void* d_ws, size_t ws_size,
                              hipStream_t stream) {
    (void)n_in;
    const float* feat = (const float*)d_in[0];
    const float* ew   = (const float*)d_in[1];
    const int*   esrc = (const int*)d_in[2];
    const int*   edst = (const int*)d_in[3];
    float*       out  = (float*)d_out;

    const int N = in_sizes[0] / D_FEAT;
    const int E = in_sizes[1];

    // cnt[N] + off[N+1] + cur[N] + dstS[E] + wS[E]
    const size_t need = ((size_t)3 * N + 1) * sizeof(unsigned)
                      + (size_t)E * (sizeof(int) + sizeof(float));

    if (ws_size >= need && N > 0) {
        unsigned* cnt  = (unsigned*)d_ws;
        unsigned* off  = cnt + N;
        unsigned* cur  = off + (N + 1);
        int*      dstS = (int*)(cur + N);
        float*    wS   = (float*)(dstS + E);

        int zb = (N + 255) / 256; if (zb > 2048) zb = 2048;
        k_zero_u32<<<zb, 256, 0, stream>>>(cnt, (size_t)N);

        int hb = (E + 255) / 256; if (hb > 4096) hb = 4096;
        k_hist<<<hb, 256, 0, stream>>>(esrc, cnt, E);

        k_scan<<<1, 1024, 0, stream>>>(cnt, off, cur, N);

        k_perm<<<hb, 256, 0, stream>>>(esrc, edst, ew, cur, dstS, wS, E);

        int gb = (N + 7) / 8; if (gb > 16384) gb = 16384;
        k_gather<<<gb, 256, 0, stream>>>(feat, wS, dstS, off, out, N);
    } else {
        int zb = (int)(((size_t)out_size + 255) / 256); if (zb > 4096) zb = 4096;
        k_zero_u32<<<zb, 256, 0, stream>>>((unsigned*)out, (size_t)out_size);

        int sb = (E + TILE - 1) / TILE; if (sb > 4096) sb = 4096;
        k_scatter_atomic<<<sb, 256, 0, stream>>>(feat, ew, esrc, edst, out, E);
    }
}